// GRCell_50689204028095
// MI455X (gfx1250) — compile-verified
//
#include <hip/hip_runtime.h>

typedef __attribute__((ext_vector_type(2))) float v2f;
typedef __attribute__((ext_vector_type(8))) float v8f;

// ---- problem constants (from reference) ----
#define DT_      1e-4f
#define TLEN     4000000L
#define L1       80          // chunk length (16*L1 must divide TLEN)
#define NCHUNK   50000       // TLEN / L1
#define GPC      16          // chunks per WMMA group (N dimension)
#define NGROUP   3125        // NCHUNK / GPC
#define SEG      100         // chunks per scan lane
#define NSEG     500         // NCHUNK / SEG
#define GPAD     81          // zero padding in front of Toeplitz table
#define GTAB     160         // padded table size (indices 0..159)
// derived physics constants: C00=200, XiCov00=2, XiCovC00=400, out scale=0.02

// ---- workspace layout (float offsets) ----
#define AL_OFF   0           // A^L (4)
#define AG_OFF   4           // (A^L)^SEG (4)
#define X0_OFF   8           // init state (2)
#define G_OFF    16          // 0.04*p00(n), n=0..79  (Toeplitz kernel)
#define Q0_OFF   96          // 0.02*p00(j)
#define Q1_OFF   176         // 0.02*p01(j)
#define W0_OFF   256         // 2*p00(L-1-i)
#define W1_OFF   336         // 2*p10(L-1-i)
#define V_OFF    512         // per-chunk offsets v_c (2*NCHUNK)
#define S_OFF    (V_OFF + 2*NCHUNK)   // per-chunk start states (2*NCHUNK)

// ================= K1: build tables from theta (1 thread) =================
__global__ void k_setup(const float* __restrict__ theta,
                        const float* __restrict__ init_state,
                        float* __restrict__ ws) {
  if (threadIdx.x != 0 || blockIdx.x != 0) return;
  double th = (double)theta[0];
  double dt = (double)DT_;
  // A = I + dt*M,  M = [[-405, th],[-th, -5]]
  double a00 = 1.0 - dt * 405.0;
  double a01 = dt * th;
  double a10 = -dt * th;
  double a11 = 1.0 - dt * 5.0;
  // p = A^n
  double p00 = 1.0, p01 = 0.0, p10 = 0.0, p11 = 1.0;
  for (int n = 0; n < L1; ++n) {
    ws[G_OFF  + n]            = (float)(0.04 * p00);
    ws[Q0_OFF + n]            = (float)(0.02 * p00);
    ws[Q1_OFF + n]            = (float)(0.02 * p01);
    ws[W0_OFF + (L1 - 1 - n)] = (float)(2.0 * p00);
    ws[W1_OFF + (L1 - 1 - n)] = (float)(2.0 * p10);
    double n00 = a00 * p00 + a01 * p10;
    double n01 = a00 * p01 + a01 * p11;
    double n10 = a10 * p00 + a11 * p10;
    double n11 = a10 * p01 + a11 * p11;
    p00 = n00; p01 = n01; p10 = n10; p11 = n11;
  }
  // p = A^L
  ws[AL_OFF + 0] = (float)p00; ws[AL_OFF + 1] = (float)p01;
  ws[AL_OFF + 2] = (float)p10; ws[AL_OFF + 3] = (float)p11;
  // Ag = (A^L)^SEG
  double m00 = 1.0, m01 = 0.0, m10 = 0.0, m11 = 1.0;
  for (int i = 0; i < SEG; ++i) {
    double n00 = p00 * m00 + p01 * m10;
    double n01 = p00 * m01 + p01 * m11;
    double n10 = p10 * m00 + p11 * m10;
    double n11 = p10 * m01 + p11 * m11;
    m00 = n00; m01 = n01; m10 = n10; m11 = n11;
  }
  ws[AG_OFF + 0] = (float)m00; ws[AG_OFF + 1] = (float)m01;
  ws[AG_OFF + 2] = (float)m10; ws[AG_OFF + 3] = (float)m11;
  ws[X0_OFF + 0] = init_state[0];
  ws[X0_OFF + 1] = init_state[1];
}

// ================= K2: per-chunk reductions v_c (400 blocks x 128 thr) ====
// each block: 125 chunks = 10000 timesteps
__global__ void k_chunk_v(const float* __restrict__ inputs,
                          float* __restrict__ ws) {
  __shared__ float sdy[10000];
  __shared__ float sw0[L1];
  __shared__ float sw1[L1];
  const int tid = threadIdx.x;
  const long base = (long)blockIdx.x * 10000;
  for (int i = tid; i < 10000; i += 128)
    sdy[i] = inputs[(base + i) * 3 + 1];
  if (tid < L1) { sw0[tid] = ws[W0_OFF + tid]; sw1[tid] = ws[W1_OFF + tid]; }
  __syncthreads();
  if (tid < 125) {
    const int c0 = tid * L1;
    float v0 = 0.f, v1 = 0.f;
    #pragma unroll 8
    for (int i = 0; i < L1; ++i) {
      float d = sdy[c0 + i];
      v0 += sw0[i] * d;
      v1 += sw1[i] * d;
    }
    long c = (long)blockIdx.x * 125 + tid;
    ws[V_OFF + 2 * c + 0] = v0;
    ws[V_OFF + 2 * c + 1] = v1;
  }
}

// ================= K3: hierarchical chunk scan (1 block x 512 thr) ========
__global__ void k_scan(float* __restrict__ ws, float* __restrict__ out) {
  __shared__ float sW[2 * NSEG];   // per-segment combined offsets
  __shared__ float sS[2 * NSEG];   // per-segment start states
  const int l = threadIdx.x;
  const float al00 = ws[AL_OFF + 0], al01 = ws[AL_OFF + 1];
  const float al10 = ws[AL_OFF + 2], al11 = ws[AL_OFF + 3];
  const float* v = ws + V_OFF;
  // Phase A: per-segment serial reduction W = sum A_L^{SEG-1-i} v_i
  if (l < NSEG) {
    float w0 = 0.f, w1 = 0.f;
    for (int i = 0; i < SEG; ++i) {
      long c = (long)l * SEG + i;
      float t0 = al00 * w0 + al01 * w1 + v[2 * c + 0];
      float t1 = al10 * w0 + al11 * w1 + v[2 * c + 1];
      w0 = t0; w1 = t1;
    }
    sW[2 * l] = w0; sW[2 * l + 1] = w1;
  }
  __syncthreads();
  // Phase B: serial scan over NSEG segments (Ag = A_L^SEG)
  if (l == 0) {
    const float ag00 = ws[AG_OFF + 0], ag01 = ws[AG_OFF + 1];
    const float ag10 = ws[AG_OFF + 2], ag11 = ws[AG_OFF + 3];
    float s0 = ws[X0_OFF + 0], s1 = ws[X0_OFF + 1];
    for (int g = 0; g < NSEG; ++g) {
      sS[2 * g] = s0; sS[2 * g + 1] = s1;
      float t0 = ag00 * s0 + ag01 * s1 + sW[2 * g + 0];
      float t1 = ag10 * s0 + ag11 * s1 + sW[2 * g + 1];
      s0 = t0; s1 = t1;
    }
    // final_state = [x_T0, x_T1, 0, 0, 0] at out[2T ..]
    out[2 * TLEN + 0] = s0;
    out[2 * TLEN + 1] = s1;
    out[2 * TLEN + 2] = 0.f;
    out[2 * TLEN + 3] = 0.f;
    out[2 * TLEN + 4] = 0.f;
  }
  __syncthreads();
  // Phase C: expand chunk start states
  if (l < NSEG) {
    float s0 = sS[2 * l], s1 = sS[2 * l + 1];
    for (int i = 0; i < SEG; ++i) {
      long c = (long)l * SEG + i;
      ws[S_OFF + 2 * c + 0] = s0;
      ws[S_OFF + 2 * c + 1] = s1;
      float t0 = al00 * s0 + al01 * s1 + v[2 * c + 0];
      float t1 = al10 * s0 + al11 * s1 + v[2 * c + 1];
      s0 = t0; s1 = t1;
    }
  }
}

// ================= K4: WMMA output pass (3125 blocks x 32 thr) ============
// one wave per group of 16 chunks (1280 timesteps):
// D(80x16) = Toeplitz(80x80) * U(80x16) via v_wmma_f32_16x16x4_f32
// Toeplitz entries are read from a zero-padded LDS table so A-tile loads are
// branch-free (no EXEC-masked DS loads in the WMMA loop).
__global__ void k_output(const float* __restrict__ inputs,
                         const float* __restrict__ ws,
                         float* __restrict__ out) {
  __shared__ float sdy[GPC * L1];   // 1280 dy values of this group
  __shared__ float sgp[GTAB];       // padded Toeplitz kernel: sgp[GPAD+n]=g(n)
  __shared__ float sq0[L1];
  __shared__ float sq1[L1];
  __shared__ float ss[2 * GPC];     // chunk start states
  __shared__ float sout[GPC * L1];  // out0 results for coalesced store
  const int lane = threadIdx.x;
  const long t0 = (long)blockIdx.x * (GPC * L1);

  for (int i = lane; i < GPC * L1; i += 32)
    sdy[i] = inputs[(t0 + i) * 3 + 1];
  for (int i = lane; i < GTAB; i += 32)
    sgp[i] = (i >= GPAD) ? ws[G_OFF + (i - GPAD)] : 0.f;
  for (int i = lane; i < L1; i += 32) {
    sq0[i] = ws[Q0_OFF + i];
    sq1[i] = ws[Q1_OFF + i];
  }
  ss[lane] = ws[S_OFF + (long)blockIdx.x * (2 * GPC) + lane];
  __syncthreads();

  const int half = lane >> 4;    // 0: lanes 0-15, 1: lanes 16-31
  const int lm   = lane & 15;
  const int k0   = half * 2;     // A-matrix K base per ISA 16x4 f32 layout

  v8f acc[5];
  #pragma unroll
  for (int rt = 0; rt < 5; ++rt)
    #pragma unroll
    for (int e = 0; e < 8; ++e) acc[rt][e] = 0.f;

  // base index into padded table for K=k0 element of row m at kb=0:
  // a.x = g(m-1-(kb*4+k0)) -> sgp[GPAD + m - 1 - kb*4 - k0]
  const int abase = GPAD + lm - 1 - k0;

  for (int kb = 0; kb < 20; ++kb) {
    // B tile 4x16: vgpr0 = rows K={0,1}, vgpr1 = rows K={2,3}, col = lm
    v2f b;
    b.x = sdy[lm * L1 + kb * 4 + half];
    b.y = sdy[lm * L1 + kb * 4 + 2 + half];
    #pragma unroll
    for (int rt = 0; rt < 5; ++rt) {
      const int ia = abase + rt * 16 - kb * 4;  // in [0, GTAB)
      v2f a;
      a.x = sgp[ia];
      a.y = sgp[ia - 1];
      acc[rt] = __builtin_amdgcn_wmma_f32_16x16x4_f32(
          false, a, false, b, (short)0, acc[rt], false, false);
    }
  }

  // add start-state term and stage results
  const float s0c = ss[2 * lm], s1c = ss[2 * lm + 1];
  #pragma unroll
  for (int rt = 0; rt < 5; ++rt)
    #pragma unroll
    for (int r = 0; r < 8; ++r) {
      const int m = rt * 16 + r + half * 8;  // D layout: vgpr r -> M=r / r+8
      sout[lm * L1 + m] = acc[rt][r] + sq0[m] * s0c + sq1[m] * s1c;
    }
  __syncthreads();

  // coalesced stores: out[2t] = 0.02*x_t[0], out[2t+1] = 0
  for (int k = 0; k < (GPC * L1) / 32; ++k) {
    const int s = lane + k * 32;
    float2 o; o.x = sout[s]; o.y = 0.f;
    *(float2*)(out + 2 * (t0 + s)) = o;
  }
}

// ================= launch =================================================
extern "C" void kernel_launch(void* const* d_in, const int* in_sizes, int n_in,
                              void* d_out, int out_size, void* d_ws, size_t ws_size,
                              hipStream_t stream) {
  (void)in_sizes; (void)n_in; (void)out_size; (void)ws_size;
  const float* inputs     = (const float*)d_in[0];
  const float* theta      = (const float*)d_in[1];
  const float* init_state = (const float*)d_in[2];
  float* out = (float*)d_out;
  float* ws  = (float*)d_ws;

  k_setup  <<<1,    1,   0, stream>>>(theta, init_state, ws);
  k_chunk_v<<<400,  128, 0, stream>>>(inputs, ws);
  k_scan   <<<1,    512, 0, stream>>>(ws, out);
  k_output <<<NGROUP, 32, 0, stream>>>(inputs, ws, out);
}